// SSMBlock_44641890074684
// MI455X (gfx1250) — compile-verified
//
#include <hip/hip_runtime.h>
#include <math.h>

typedef float v2f __attribute__((ext_vector_type(2)));
typedef float v8f __attribute__((ext_vector_type(8)));

#define LROW 8192
#define NSTATE 64
#define PI_F 3.14159265358979323846f

// ---- workspace layout (in floats) ----
// F128Q / IF128Q: per row (128), per K-pair j (64): [r(2j), r(2j+1), i(2j), i(2j+1)]
#define OFF_F128Q   0          // 32768
#define OFF_IF128Q  32768      // 32768
// G64P* : pair-interleaved B layout: pos = (row>>1)*128 + 2*col + (row&1)
#define OFF_G64PR   65536      // 4096
#define OFF_G64PI   69632      // 4096
#define OFF_IG64PR  73728      // 4096
#define OFF_IG64PI  77824      // 4096
// twiddles, interleaved complex: [idx*2]=r,[idx*2+1]=i ; idx = k2*64+n1
#define OFF_WF      81920      // 16384
#define OFF_WI      98304      // 16384
// K_hat per layer, interleaved complex, storage idx s = k2*64+k1 (k = 128*k1+k2), scaled 1/L
#define OFF_KHAT    114688     // 4 * 16384
#define WS_FLOATS   (OFF_KHAT + 4*16384)

static __device__ inline v8f wmma4(v2f a, v2f b, v8f c) {
  // D = A(16x4) * B(4x16) + C(16x16), fp32
  return __builtin_amdgcn_wmma_f32_16x16x4_f32(false, a, false, b, (short)0, c,
                                               false, false);
}

// pair-interleaved LDS index: rows 2j,2j+1 of a [128][64] matrix stored adjacent
static __device__ inline int pidx(int row, int col) {
  return ((row >> 1) << 7) + (col << 1) + (row & 1);
}

// ------------------------------------------------------------------
// Build DFT matrices + twiddle tables (16384 threads)
// ------------------------------------------------------------------
__global__ void ssm_gen_tables(float* __restrict__ ws) {
  int idx = blockIdx.x * blockDim.x + threadIdx.x; // 0..16383
  {
    int k2 = idx >> 7, n2 = idx & 127;             // row k2, col n2
    int t = (k2 * n2) & 127;
    float s, c;
    sincosf(-2.f * PI_F * (float)t / 128.f, &s, &c);
    int pos = k2 * 256 + (n2 >> 1) * 4 + (n2 & 1); // A-frag-friendly quad layout
    ws[OFF_F128Q + pos]      = c;
    ws[OFF_F128Q + pos + 2]  = s;
    ws[OFF_IF128Q + pos]     = c;
    ws[OFF_IF128Q + pos + 2] = -s;
  }
  if (idx < 8192) {
    int k2 = idx >> 6, n1 = idx & 63;
    int t = (n1 * k2) & 8191;
    float s, c;
    sincosf(-2.f * PI_F * (float)t / 8192.f, &s, &c);
    ws[OFF_WF + 2 * idx]     = c;  ws[OFF_WF + 2 * idx + 1] = s;
    ws[OFF_WI + 2 * idx]     = c;  ws[OFF_WI + 2 * idx + 1] = -s;
  }
  if (idx < 4096) {
    int n1 = idx >> 6, k1 = idx & 63;              // row n1, col k1
    int t = (n1 * k1) & 63;
    float s, c;
    sincosf(-2.f * PI_F * (float)t / 64.f, &s, &c);
    int pos = (n1 >> 1) * 128 + 2 * k1 + (n1 & 1); // B-frag-friendly pair layout
    ws[OFF_G64PR + pos]  = c;  ws[OFF_G64PI + pos]  = s;
    ws[OFF_IG64PR + pos] = c;  ws[OFF_IG64PI + pos] = -s;
  }
}

// ------------------------------------------------------------------
// DPLR generating-function kernel K_hat, permuted to [k2][k1] layout,
// interleaved complex, pre-scaled by 1/L.  grid = (32,4), 256 thr.
// ------------------------------------------------------------------
__global__ void ssm_gen_khat(float* __restrict__ ws,
                             const float* __restrict__ Lr,
                             const float* __restrict__ Li,
                             const float* __restrict__ Pv,
                             const float* __restrict__ Bv,
                             const float* __restrict__ Ct,
                             const float* __restrict__ stepv) {
  int layer = blockIdx.y;
  int s = blockIdx.x * blockDim.x + threadIdx.x; // 0..8191 storage index
  int k2 = s >> 6, k1 = s & 63;
  int kf = 128 * k1 + k2;                        // natural FFT bin

  float osin, ocos;
  sincosf(-2.f * PI_F * (float)kf / 8192.f, &osin, &ocos);
  float pr = 1.f + ocos, pi = osin;              // 1+Omega
  float mr = 1.f - ocos, mi = -osin;             // 1-Omega
  float pd = pr * pr + pi * pi;
  float ipr = pr / pd, ipi = -pi / pd;           // 1/(1+Omega)
  float cr = 2.f * ipr, ci = 2.f * ipi;          // c
  float sc = 2.f / stepv[layer];
  float gr = sc * (mr * ipr - mi * ipi);
  float gi = sc * (mr * ipi + mi * ipr);         // g

  const float* lr = Lr + layer * NSTATE;
  const float* li = Li + layer * NSTATE;
  const float* pp = Pv + layer * NSTATE;
  const float* bb = Bv + layer * NSTATE;
  const float* ct = Ct + layer * NSTATE;

  float k00r = 0, k00i = 0, k01r = 0, k01i = 0;
  float k10r = 0, k10i = 0, k11r = 0, k11i = 0;
  for (int n = 0; n < NSTATE; ++n) {
    float dr = gr - lr[n], di = gi - li[n];
    float dd = dr * dr + di * di;
    float ir = dr / dd, ii = -di / dd;           // 1/(g - Lambda_n)
    float b = bb[n], p = pp[n], c = ct[n];
    float w00 = c * b, w01 = c * p, w10 = p * b, w11 = p * p;
    k00r += w00 * ir; k00i += w00 * ii;
    k01r += w01 * ir; k01i += w01 * ii;
    k10r += w10 * ir; k10i += w10 * ii;
    k11r += w11 * ir; k11i += w11 * ii;
  }
  float qr = k01r * k10r - k01i * k10i;
  float qi = k01r * k10i + k01i * k10r;
  float er = 1.f + k11r, ei = k11i;
  float ed = er * er + ei * ei;
  float ier = er / ed, iei = -ei / ed;
  float tr = qr * ier - qi * iei;
  float ti = qr * iei + qi * ier;
  float kr = k00r - tr, ki = k00i - ti;
  float outr = cr * kr - ci * ki;
  float outi = cr * ki + ci * kr;
  const float scale = 1.f / 8192.f;              // IFFT normalization folded in
  ws[OFF_KHAT + layer * 16384 + 2 * s]     = outr * scale;
  ws[OFF_KHAT + layer * 16384 + 2 * s + 1] = outi * scale;
}

// ------------------------------------------------------------------
// Main: one workgroup (256 thr = 8 wave32) per row; 4 layers in 96 KB LDS.
// Four-step FFT (128x64 split), all matmuls on v_wmma_f32_16x16x4_f32.
// A wave owns one 16-row M-slice in every stage, so stage2 (xK_hat) and
// inv-stage1 operate IN PLACE on Y with no barrier and no Z buffer.
// xb and Y are stored pair-interleaved by row so every LDS fragment is a
// single b64 load (B) or a single ds_load_2addr pair (A).
// ------------------------------------------------------------------
__global__ __launch_bounds__(256) void ssm_rows(const float* __restrict__ uin,
                                                float* __restrict__ uout,
                                                const float* __restrict__ ws) {
  extern __shared__ float smem[];
  float* xb = smem;           // [128][64] real, pair-interleaved
  float* Yr = smem + 8192;    // [128][64] pair-interleaved
  float* Yi = smem + 16384;

  const int tid  = threadIdx.x;
  const int wave = tid >> 5;
  const int lane = tid & 31;
  const int l16  = lane & 15;
  const int kSel = (lane >> 4) << 1;   // 0 or 2 : K-pair selected by half-wave
  const int mHi  = (lane >> 4) << 3;   // 0 or 8 : C-row offset by half-wave
  const int mB   = wave << 4;          // this wave's M-slice (all stages M=128)

  const size_t rowOff = (size_t)blockIdx.x * LROW;
  const float4* src = (const float4*)(uin + rowOff);
  for (int i = tid; i < LROW / 4; i += 256) {
    float4 v = src[i];
    int n2 = i >> 4, n1 = (i & 15) << 2;
    int base = pidx(n2, n1);
    xb[base] = v.x; xb[base + 2] = v.y; xb[base + 4] = v.z; xb[base + 6] = v.w;
  }
  __syncthreads();

  const float* F128Q  = ws + OFF_F128Q;
  const float* IF128Q = ws + OFF_IF128Q;
  const float* GPr    = ws + OFF_G64PR;
  const float* GPi    = ws + OFF_G64PI;
  const float* iGPr   = ws + OFF_IG64PR;
  const float* iGPi   = ws + OFF_IG64PI;
  const float* Wf     = ws + OFF_WF;
  const float* Wi     = ws + OFF_WI;

  for (int layer = 0; layer < 4; ++layer) {
    const float* kh = ws + OFF_KHAT + layer * 16384;

    // ---- forward stage 1: Y = F128 * X, fused twiddle Wf (writes own rows) ----
    {
      v8f cr[4] = {}, ci[4] = {};
      for (int kk = 0; kk < 128; kk += 4) {
        float4 aq = *(const float4*)(F128Q + (mB + l16) * 256 + (kk + kSel) * 2);
        v2f ar; ar.x = aq.x; ar.y = aq.y;
        v2f ai; ai.x = aq.z; ai.y = aq.w;
        int jb = ((kk + kSel) >> 1) << 7;
#pragma unroll
        for (int t = 0; t < 4; ++t) {
          v2f b = *(const v2f*)(xb + jb + (((t << 4) + l16) << 1));
          cr[t] = wmma4(ar, b, cr[t]);
          ci[t] = wmma4(ai, b, ci[t]);
        }
      }
#pragma unroll
      for (int t = 0; t < 4; ++t) {
#pragma unroll
        for (int v = 0; v < 8; ++v) {
          int row = mB + v + mHi, col = (t << 4) + l16;
          v2f w = *(const v2f*)(Wf + ((row << 6) + col) * 2);
          int p = pidx(row, col);
          float a = cr[t][v], b = ci[t][v];
          Yr[p] = a * w.x - b * w.y;
          Yi[p] = a * w.y + b * w.x;
        }
      }
    }
    // no barrier: stage2 reads only this wave's rows of Y

    // ---- forward stage 2 (in place): Y = (Y * G64) * K_hat ----
    {
      v8f zr[4] = {}, zi[4] = {};
      for (int kk = 0; kk < 64; kk += 4) {
        int p0 = pidx(mB + l16, kk + kSel);
        v2f ar; ar.x = Yr[p0]; ar.y = Yr[p0 + 2];
        v2f ai; ai.x = Yi[p0]; ai.y = Yi[p0 + 2];
        int jrow = ((kk + kSel) >> 1) << 7;
#pragma unroll
        for (int t = 0; t < 4; ++t) {
          int cc = ((t << 4) + l16) << 1;
          v2f br = *(const v2f*)(GPr + jrow + cc);
          v2f bi = *(const v2f*)(GPi + jrow + cc);
          v2f nbi; nbi.x = -bi.x; nbi.y = -bi.y;
          zr[t] = wmma4(ar, br, zr[t]);  zr[t] = wmma4(ai, nbi, zr[t]);
          zi[t] = wmma4(ar, bi, zi[t]);  zi[t] = wmma4(ai, br, zi[t]);
        }
      }
#pragma unroll
      for (int t = 0; t < 4; ++t) {
#pragma unroll
        for (int v = 0; v < 8; ++v) {
          int row = mB + v + mHi, col = (t << 4) + l16;
          v2f k = *(const v2f*)(kh + ((row << 6) + col) * 2);
          int p = pidx(row, col);
          float a = zr[t][v], b = zi[t][v];
          Yr[p] = a * k.x - b * k.y;
          Yi[p] = a * k.y + b * k.x;
        }
      }
    }
    // no barrier: inv stage1 reads only this wave's rows of Y

    // ---- inverse stage 1 (in place): Y = (Y * iG64), fused twiddle Wi ----
    {
      v8f yr[4] = {}, yi[4] = {};
      for (int kk = 0; kk < 64; kk += 4) {
        int p0 = pidx(mB + l16, kk + kSel);
        v2f ar; ar.x = Yr[p0]; ar.y = Yr[p0 + 2];
        v2f ai; ai.x = Yi[p0]; ai.y = Yi[p0 + 2];
        int jrow = ((kk + kSel) >> 1) << 7;
#pragma unroll
        for (int t = 0; t < 4; ++t) {
          int cc = ((t << 4) + l16) << 1;
          v2f br = *(const v2f*)(iGPr + jrow + cc);
          v2f bi = *(const v2f*)(iGPi + jrow + cc);
          v2f nbi; nbi.x = -bi.x; nbi.y = -bi.y;
          yr[t] = wmma4(ar, br, yr[t]);  yr[t] = wmma4(ai, nbi, yr[t]);
          yi[t] = wmma4(ar, bi, yi[t]);  yi[t] = wmma4(ai, br, yi[t]);
        }
      }
#pragma unroll
      for (int t = 0; t < 4; ++t) {
#pragma unroll
        for (int v = 0; v < 8; ++v) {
          int row = mB + v + mHi, col = (t << 4) + l16;
          v2f w = *(const v2f*)(Wi + ((row << 6) + col) * 2);
          int p = pidx(row, col);
          float a = yr[t][v], b = yi[t][v];
          Yr[p] = a * w.x - b * w.y;
          Yi[p] = a * w.y + b * w.x;
        }
      }
    }
    __syncthreads();   // inv stage 2 reads ALL rows of Y

    // ---- inverse stage 2: x = gelu(Re(iF128 * Y) + x) ----
    {
      v8f acc[4] = {};
      for (int kk = 0; kk < 128; kk += 4) {
        float4 aq = *(const float4*)(IF128Q + (mB + l16) * 256 + (kk + kSel) * 2);
        v2f ar; ar.x = aq.x; ar.y = aq.y;
        v2f ai; ai.x = aq.z; ai.y = aq.w;
        int jb = ((kk + kSel) >> 1) << 7;
#pragma unroll
        for (int t = 0; t < 4; ++t) {
          int cc = ((t << 4) + l16) << 1;
          v2f br  = *(const v2f*)(Yr + jb + cc);
          v2f byi = *(const v2f*)(Yi + jb + cc);
          v2f nbi; nbi.x = -byi.x; nbi.y = -byi.y;
          acc[t] = wmma4(ar, br, acc[t]);
          acc[t] = wmma4(ai, nbi, acc[t]);
        }
      }
#pragma unroll
      for (int t = 0; t < 4; ++t) {
#pragma unroll
        for (int v = 0; v < 8; ++v) {
          int row = mB + v + mHi, col = (t << 4) + l16;  // row = n2, col = n1
          int p = pidx(row, col);
          float y = acc[t][v] + xb[p];
          xb[p] = 0.5f * y * (1.f + erff(y * 0.70710678118654752f));
        }
      }
    }
    __syncthreads();   // next layer's stage 1 reads ALL rows of xb
  }

  float4* dst = (float4*)(uout + rowOff);
  for (int i = tid; i < LROW / 4; i += 256) {
    int n2 = i >> 4, n1 = (i & 15) << 2;
    int base = pidx(n2, n1);
    float4 o;
    o.x = xb[base]; o.y = xb[base + 2]; o.z = xb[base + 4]; o.w = xb[base + 6];
    dst[i] = o;
  }
}

// ------------------------------------------------------------------
extern "C" void kernel_launch(void* const* d_in, const int* in_sizes, int n_in,
                              void* d_out, int out_size, void* d_ws, size_t ws_size,
                              hipStream_t stream) {
  const float* u    = (const float*)d_in[0];
  const float* Lr   = (const float*)d_in[1];
  const float* Li   = (const float*)d_in[2];
  const float* Pv   = (const float*)d_in[3];
  const float* Bv   = (const float*)d_in[4];
  const float* Ct   = (const float*)d_in[5];
  const float* step = (const float*)d_in[6];
  float* out = (float*)d_out;
  float* ws  = (float*)d_ws;

  int rows = in_sizes[0] / LROW;   // 8*512 = 4096

  ssm_gen_tables<<<64, 256, 0, stream>>>(ws);
  ssm_gen_khat<<<dim3(32, 4), 256, 0, stream>>>(ws, Lr, Li, Pv, Bv, Ct, step);
  ssm_rows<<<rows, 256, 24576 * sizeof(float), stream>>>(u, out, ws);
}